// SRU_18081812316549
// MI455X (gfx1250) — compile-verified
//
#include <hip/hip_runtime.h>
#include <hip/hip_bf16.h>
#include <cstdint>

typedef float v2f __attribute__((ext_vector_type(2)));
typedef float v8f __attribute__((ext_vector_type(8)));
typedef int   v4i_gcc __attribute__((vector_size(16)));            // matches builtin's V4i
typedef v4i_gcc __attribute__((address_space(1)))* glb_v4i_ptr;    // global int4*
typedef v4i_gcc __attribute__((address_space(3)))* lds_v4i_ptr;    // LDS int4*

// Problem constants
#define TT   4096
#define BB   8
#define DD   512
#define HH   512
#define N3   1536            // 3*H
#define MM   (TT * BB)       // 32768 GEMM rows
#define CH   (BB * HH)       // 4096 scan channels
#define NCHK 64              // chunks over T
#define CLEN 64              // chunk length (NCHK*CLEN == TT)

// GEMM tiling: macrotile 64(M) x 128(N), 8 waves in 2(M) x 4(N), wave tile 32x32
#define BM 64
#define BN 128
#define TK 32
#define LDP 36               // padded LDS row stride (floats): 144B -> 16B aligned, bank-friendly
#define NKCHUNK (DD / TK)    // 16

#if defined(__has_builtin)
#if __has_builtin(__builtin_amdgcn_global_load_async_to_lds_b128) && \
    __has_builtin(__builtin_amdgcn_s_wait_asynccnt)
#define CDNA5_ASYNC 1
#endif
#endif
#ifndef CDNA5_ASYNC
#define CDNA5_ASYNC 0
#endif

__device__ __forceinline__ float sigmoidf_fast(float x) {
    return 1.0f / (1.0f + __expf(-x));
}
__device__ __forceinline__ float tanhf_fast(float x) {
    return 1.0f - 2.0f / (__expf(2.0f * x) + 1.0f);
}

#if CDNA5_ASYNC
__device__ __forceinline__ void async_b128(const float* g, const float* l) {
    __builtin_amdgcn_global_load_async_to_lds_b128(
        (glb_v4i_ptr)(uintptr_t)g,
        (lds_v4i_ptr)(uint32_t)(uintptr_t)l,
        0, 0);
}
#endif

// -------------------------------------------------------------------------
// GEMM: U[M, N3] = Xin[M, 512] * W[N3, 512]^T  via V_WMMA_F32_16X16X4_F32
// -------------------------------------------------------------------------
__global__ __launch_bounds__(256) void sru_gemm_f32(const float* __restrict__ Xin,
                                                    const float* __restrict__ W,
                                                    float* __restrict__ U) {
#if CDNA5_ASYNC
    __shared__ float As[2][BM * LDP];
    __shared__ float Bs[2][BN * LDP];
#else
    __shared__ float As[1][BM * LDP];
    __shared__ float Bs[1][BN * LDP];
#endif

    const int m0   = blockIdx.x * BM;
    const int n0   = blockIdx.y * BN;
    const int tid  = threadIdx.x;
    const int wave = tid >> 5;
    const int lane = tid & 31;
    const int wmr  = wave & 1;        // 2 waves along M
    const int wnc  = wave >> 1;       // 4 waves along N
    const int lrow  = lane & 15;
    const int khalf = (lane < 16) ? 0 : 2;

    v8f acc00 = {}, acc01 = {}, acc10 = {}, acc11 = {};

#if CDNA5_ASYNC
    // Issue 6 async-to-LDS B128 ops per thread for one K-chunk (2 for A, 4 for B).
    auto issue_loads = [&](int buf, int k0) {
        #pragma unroll
        for (int i = 0; i < 2; ++i) {
            const int linear = tid + i * 256;
            const int r  = linear >> 3;          // 0..63
            const int cb = (linear & 7) << 2;
            async_b128(Xin + (size_t)(m0 + r) * DD + k0 + cb, &As[buf][r * LDP + cb]);
        }
        #pragma unroll
        for (int i = 0; i < 4; ++i) {
            const int linear = tid + i * 256;
            const int r  = linear >> 3;          // 0..127
            const int cb = (linear & 7) << 2;
            async_b128(W + (size_t)(n0 + r) * DD + k0 + cb, &Bs[buf][r * LDP + cb]);
        }
    };
#else
    auto stage_loads = [&](int k0) {
        #pragma unroll
        for (int i = 0; i < 2; ++i) {
            const int linear = tid + i * 256;
            const int r  = linear >> 3;
            const int cb = (linear & 7) << 2;
            const float4 v = *(const float4*)(Xin + (size_t)(m0 + r) * DD + k0 + cb);
            float* dst = &As[0][r * LDP + cb];
            dst[0] = v.x; dst[1] = v.y; dst[2] = v.z; dst[3] = v.w;
        }
        #pragma unroll
        for (int i = 0; i < 4; ++i) {
            const int linear = tid + i * 256;
            const int r  = linear >> 3;
            const int cb = (linear & 7) << 2;
            const float4 v = *(const float4*)(W + (size_t)(n0 + r) * DD + k0 + cb);
            float* dst = &Bs[0][r * LDP + cb];
            dst[0] = v.x; dst[1] = v.y; dst[2] = v.z; dst[3] = v.w;
        }
    };
#endif

    // 8 K-steps x 4 independent WMMAs per chunk; 2 A-frags + 2 B-frags feed 4 WMMAs.
    auto compute = [&](int buf) {
        #pragma unroll
        for (int kk = 0; kk < TK; kk += 4) {
            v2f a0, a1, b0, b1;
            const float* ap0 = &As[buf][(wmr * 32 + lrow) * LDP + kk + khalf];
            const float* ap1 = ap0 + 16 * LDP;
            a0.x = ap0[0]; a0.y = ap0[1];
            a1.x = ap1[0]; a1.y = ap1[1];
            const float* bp0 = &Bs[buf][(wnc * 32 + lrow) * LDP + kk + khalf];
            const float* bp1 = bp0 + 16 * LDP;
            b0.x = bp0[0]; b0.y = bp0[1];
            b1.x = bp1[0]; b1.y = bp1[1];
            acc00 = __builtin_amdgcn_wmma_f32_16x16x4_f32(false, a0, false, b0, (short)0, acc00, false, false);
            acc01 = __builtin_amdgcn_wmma_f32_16x16x4_f32(false, a0, false, b1, (short)0, acc01, false, false);
            acc10 = __builtin_amdgcn_wmma_f32_16x16x4_f32(false, a1, false, b0, (short)0, acc10, false, false);
            acc11 = __builtin_amdgcn_wmma_f32_16x16x4_f32(false, a1, false, b1, (short)0, acc11, false, false);
        }
    };

#if CDNA5_ASYNC
    issue_loads(0, 0);
    issue_loads(1, TK);
    for (int c = 0; c < NKCHUNK; ++c) {
        // 6 async ops per chunk per wave, in-order completion: allowing 6 outstanding
        // means chunk c has fully landed while chunk c+1 may still be in flight.
        if (c == NKCHUNK - 1) __builtin_amdgcn_s_wait_asynccnt(0);
        else                  __builtin_amdgcn_s_wait_asynccnt(6);
        __syncthreads();
        compute(c & 1);
        __syncthreads();                 // everyone done reading this buffer
        if (c + 2 < NKCHUNK) issue_loads(c & 1, (c + 2) * TK);
    }
#else
    for (int c = 0; c < NKCHUNK; ++c) {
        stage_loads(c * TK);
        __syncthreads();
        compute(0);
        __syncthreads();
    }
#endif

    // D layout: VGPR j -> M = j (lanes 0-15) / 8+j (lanes 16-31); N = lane&15
    const int cb0 = n0 + wnc * 32 + lrow;
    const int mb  = m0 + wmr * 32 + ((lane < 16) ? 0 : 8);
    #pragma unroll
    for (int j = 0; j < 8; ++j) {
        U[(size_t)(mb + j)      * N3 + cb0]      = acc00[j];
        U[(size_t)(mb + j)      * N3 + cb0 + 16] = acc01[j];
        U[(size_t)(mb + 16 + j) * N3 + cb0]      = acc10[j];
        U[(size_t)(mb + 16 + j) * N3 + cb0 + 16] = acc11[j];
    }
}

// -------------------------------------------------------------------------
// Scan pass A: per-(channel, chunk) summaries: P = prod f, Z = chunk scan from 0
// -------------------------------------------------------------------------
__global__ __launch_bounds__(256) void sru_scan_chunks(const float* __restrict__ U,
                                                       const float* __restrict__ bias_f,
                                                       float* __restrict__ Pbuf,
                                                       float* __restrict__ Zbuf) {
    const int gid = blockIdx.x * blockDim.x + threadIdx.x;
    const int ch    = gid & (CH - 1);
    const int chunk = gid >> 12;
    const int bi = ch >> 9;
    const int hi = ch & (HH - 1);
    const float bf = bias_f[hi];

    float P = 1.0f, z = 0.0f;
    const int tbase = chunk * CLEN;
    #pragma unroll 4
    for (int i = 0; i < CLEN; ++i) {
        const size_t row = (size_t)((tbase + i) * BB + bi) * N3;
        const float xt = U[row + hi];
        const float f  = sigmoidf_fast(U[row + HH + hi] + bf);
        z = f * z + (1.0f - f) * xt;
        P *= f;
    }
    Pbuf[chunk * CH + ch] = P;
    Zbuf[chunk * CH + ch] = z;
}

// -------------------------------------------------------------------------
// Scan pass B: sequential combine across chunk summaries; emits carries.
// -------------------------------------------------------------------------
__global__ __launch_bounds__(256) void sru_scan_carry(const float* __restrict__ Pbuf,
                                                      const float* __restrict__ Zbuf,
                                                      float* __restrict__ Cin,
                                                      float* __restrict__ c_out,
                                                      int write_c) {
    const int ch = blockIdx.x * blockDim.x + threadIdx.x;
    float c = 0.0f;
    #pragma unroll 8
    for (int j = 0; j < NCHK; ++j) {
        Cin[j * CH + ch] = c;
        c = Pbuf[j * CH + ch] * c + Zbuf[j * CH + ch];
    }
    if (write_c) c_out[ch] = c;
}

// -------------------------------------------------------------------------
// Scan pass C: replay each chunk from its carry, emit h (hout may alias xin).
// -------------------------------------------------------------------------
__global__ __launch_bounds__(256) void sru_scan_emit(const float* __restrict__ U,
                                                     const float* __restrict__ bias,
                                                     const float* xin,
                                                     const float* __restrict__ Cin,
                                                     float* hout) {
    const int gid = blockIdx.x * blockDim.x + threadIdx.x;
    const int ch    = gid & (CH - 1);
    const int chunk = gid >> 12;
    const int bi = ch >> 9;
    const int hi = ch & (HH - 1);
    const float bf = bias[hi];
    const float br = bias[HH + hi];

    float c = Cin[chunk * CH + ch];
    const int tbase = chunk * CLEN;
    #pragma unroll 4
    for (int i = 0; i < CLEN; ++i) {
        const int t = tbase + i;
        const size_t urow = (size_t)(t * BB + bi) * N3;
        const float xt = U[urow + hi];
        const float f  = sigmoidf_fast(U[urow + HH + hi] + bf);
        const float r  = sigmoidf_fast(U[urow + 2 * HH + hi] + br);
        c = f * c + (1.0f - f) * xt;
        const size_t xrow = (size_t)(t * BB + bi) * HH + hi;
        const float xv = xin[xrow];
        hout[xrow] = r * tanhf_fast(c) + (1.0f - r) * xv;
    }
}

extern "C" void kernel_launch(void* const* d_in, const int* in_sizes, int n_in,
                              void* d_out, int out_size, void* d_ws, size_t ws_size,
                              hipStream_t stream) {
    const float* x  = (const float*)d_in[0];
    const float* W0 = (const float*)d_in[1];
    const float* b0 = (const float*)d_in[2];
    const float* W1 = (const float*)d_in[3];
    const float* b1 = (const float*)d_in[4];
    float* out = (float*)d_out;

    char* ws = (char*)d_ws;
    float* U    = (float*)(ws);                                  // 192 MB
    float* Pbuf = (float*)(ws + (size_t)MM * N3 * 4);            // 1 MB
    float* Zbuf = Pbuf + NCHK * CH;                              // 1 MB
    float* Cin  = Zbuf + NCHK * CH;                              // 1 MB

    float* h0    = out;                 // layer-0 h lives in d_out, overwritten in place by layer 1
    float* c_out = out + (size_t)TT * BB * HH;

    const dim3 gemm_grid(MM / BM, N3 / BN);       // (512, 12)
    const int scan_blocks  = (NCHK * CH) / 256;   // 1024
    const int carry_blocks = CH / 256;            // 16

    // ---- Layer 0 ----
    sru_gemm_f32<<<gemm_grid, 256, 0, stream>>>(x, W0, U);
    sru_scan_chunks<<<scan_blocks, 256, 0, stream>>>(U, b0, Pbuf, Zbuf);
    sru_scan_carry<<<carry_blocks, 256, 0, stream>>>(Pbuf, Zbuf, Cin, c_out, /*write_c=*/0);
    sru_scan_emit<<<scan_blocks, 256, 0, stream>>>(U, b0, x, Cin, h0);

    // ---- Layer 1 (c0 resets to zero per the reference) ----
    sru_gemm_f32<<<gemm_grid, 256, 0, stream>>>(h0, W1, U);
    sru_scan_chunks<<<scan_blocks, 256, 0, stream>>>(U, b1, Pbuf, Zbuf);
    sru_scan_carry<<<carry_blocks, 256, 0, stream>>>(Pbuf, Zbuf, Cin, c_out, /*write_c=*/1);
    sru_scan_emit<<<scan_blocks, 256, 0, stream>>>(U, b1, h0, Cin, out);
}